// PillarFeatureExtraction_2989297238034
// MI455X (gfx1250) — compile-verified
//
#include <hip/hip_runtime.h>

typedef __attribute__((ext_vector_type(2))) float v2f;
typedef __attribute__((ext_vector_type(4))) float v4f;
typedef __attribute__((ext_vector_type(8))) float v8f;

#define GRID_XD 432
#define GRID_YD 496
#define NPTS    32
#define CCH     64
#define KPAD    12
#define VXC     0.16f
#define XOFF    0.08f
#define YOFF   (-39.60f)
#define ZCONST (-1.0f)
#define BN_EPS  1e-3f

__device__ __forceinline__ float waveSum(float v) {
#pragma unroll
    for (int off = 16; off; off >>= 1) v += __shfl_xor(v, off, 32);
    return v;
}

// Per-point 10-dim pillar feature, masked.
__device__ __forceinline__ void computeFeats(float4 pt, float mx, float my, float mz,
                                             float ccx, float ccy, float m, float f[10]) {
    f[0] = (pt.x - ccx) * m;
    f[1] = (pt.y - ccy) * m;
    f[2] = pt.z * m;
    f[3] = pt.w * m;
    f[4] = (pt.x - mx) * m;
    f[5] = (pt.y - my) * m;
    f[6] = (pt.z - mz) * m;
    f[7] = f[0];
    f[8] = f[1];
    f[9] = ZCONST * m;
}

// ---- Kernel 1: zero output grid (non-temporal; grid > L2) + ws accumulators
__global__ void zero_kernel(float* __restrict__ out, long n, float* __restrict__ ws) {
    long i = (long)blockIdx.x * blockDim.x + threadIdx.x;
    long stride = (long)gridDim.x * blockDim.x;
    long n4 = n >> 2;
    v4f* o4 = (v4f*)out;
    v4f z = {0.f, 0.f, 0.f, 0.f};
    for (long j = i; j < n4; j += stride) __builtin_nontemporal_store(z, &o4[j]);
    for (long j = (n4 << 2) + i; j < n; j += stride) out[j] = 0.f;
    if (blockIdx.x == 0 && threadIdx.x < 256) ws[threadIdx.x] = 0.f;
}

// ---- Kernel 2: feature moments M1 (10) + M2 upper-tri (55) -----------------
// Persistent waves: accumulate in registers across pillars; one 65-way wave
// reduction at the end (instead of per-pillar) -> memory-bound pass.
__global__ void __launch_bounds__(256)
moments_kernel(const float* __restrict__ pillars, const int* __restrict__ cxp,
               const int* __restrict__ cyp, const int* __restrict__ npp,
               float* __restrict__ mom, int P) {
    int tid = threadIdx.x;
    int wave = tid >> 5, lane = tid & 31;
    int gwave = blockIdx.x * 8 + wave;
    int nwaves = gridDim.x * 8;

    float acc[65];
#pragma unroll
    for (int t = 0; t < 65; t++) acc[t] = 0.f;

    for (int p = gwave; p < P; p += nwaves) {
        float4 pt = ((const float4*)pillars)[(long)p * NPTS + lane];
        float sx = waveSum(pt.x), sy = waveSum(pt.y), sz = waveSum(pt.z);
        int npv = npp[p];
        float inv = 1.f / (float)npv;
        float ccx = (float)cxp[p] * VXC + XOFF;
        float ccy = (float)cyp[p] * VXC + YOFF;
        float m = (lane < npv) ? 1.f : 0.f;
        float f[10];
        computeFeats(pt, sx * inv, sy * inv, sz * inv, ccx, ccy, m, f);
#pragma unroll
        for (int i = 0; i < 10; i++) acc[i] += f[i];
        int idx = 10;
#pragma unroll
        for (int i = 0; i < 10; i++) {
#pragma unroll
            for (int j = i; j < 10; j++) {
                acc[idx] += f[i] * f[j];
                idx++;
            }
        }
    }

    __shared__ float blk[65];
    if (tid < 65) blk[tid] = 0.f;
    __syncthreads();
#pragma unroll
    for (int t = 0; t < 65; t++) {
        float v = waveSum(acc[t]);
        if (lane == 0) atomicAdd(&blk[t], v);
    }
    __syncthreads();
    if (tid < 65) atomicAdd(&mom[tid], blk[tid]);
}

// ---- Kernel 3: fold moments through W -> per-channel BN scale/shift --------
__global__ void finalize_bn(const float* __restrict__ mom, const float* __restrict__ W,
                            const float* __restrict__ gamma, const float* __restrict__ beta,
                            float* __restrict__ scaleArr, float* __restrict__ shiftArr,
                            float invPN) {
    __shared__ float m1[10];
    __shared__ float m2[55];
    int c = threadIdx.x;  // 64 threads
    if (c < 10) m1[c] = mom[c] * invPN;
    if (c < 55) m2[c] = mom[10 + c] * invPN;
    __syncthreads();
    float w[10];
#pragma unroll
    for (int k = 0; k < 10; k++) w[k] = W[c * 10 + k];
    float mu = 0.f;
#pragma unroll
    for (int k = 0; k < 10; k++) mu += w[k] * m1[k];
    float e2 = 0.f;
    int idx = 0;
#pragma unroll
    for (int i = 0; i < 10; i++) {
#pragma unroll
        for (int j = i; j < 10; j++) {
            float t = w[i] * w[j] * m2[idx];
            e2 += (i == j) ? t : 2.f * t;
            idx++;
        }
    }
    float var = e2 - mu * mu;
    float sc = gamma[c] * rsqrtf(var + BN_EPS);
    scaleArr[c] = sc;
    shiftArr[c] = beta[c] - mu * sc;
}

// ---- Kernel 4: feats -> WMMA f32 GEMM -> BN affine -> ReLU -> max -> scatter
__global__ void __launch_bounds__(256)
pillar_wmma_kernel(const float* __restrict__ pillars, const float* __restrict__ W,
                   const int* __restrict__ cxp, const int* __restrict__ cyp,
                   const int* __restrict__ cbp, const int* __restrict__ npp,
                   const float* __restrict__ scaleArr, const float* __restrict__ shiftArr,
                   float* __restrict__ out, int P) {
    __shared__ float Wl[CCH][KPAD];        // 64 x 12 (K padded, zeros at k>=10)
    __shared__ float Fl[8][NPTS][KPAD];    // per-wave pillar features

    int tid = threadIdx.x;
    for (int e = tid; e < CCH * KPAD; e += 256) {
        int c = e / KPAD, k = e % KPAD;
        Wl[c][k] = (k < 10) ? W[c * 10 + k] : 0.f;
    }

    int wave = tid >> 5, lane = tid & 31;
    int p = blockIdx.x * 8 + wave;
    int pc = p < P ? p : P - 1;

    float4 pt = ((const float4*)pillars)[(long)pc * NPTS + lane];
    float sx = waveSum(pt.x), sy = waveSum(pt.y), sz = waveSum(pt.z);
    int npv = npp[pc];
    float inv = 1.f / (float)npv;
    float ccx = (float)cxp[pc] * VXC + XOFF;
    float ccy = (float)cyp[pc] * VXC + YOFF;
    float m = (lane < npv) ? 1.f : 0.f;
    float f[10];
    computeFeats(pt, sx * inv, sy * inv, sz * inv, ccx, ccy, m, f);
#pragma unroll
    for (int k = 0; k < 10; k++) Fl[wave][lane][k] = f[k];
    Fl[wave][lane][10] = 0.f;
    Fl[wave][lane][11] = 0.f;
    __syncthreads();

    // WMMA phase. A-matrix 16x4 f32 layout: lanes 0-15 hold K=0,1 of row M=lane;
    // lanes 16-31 hold K=2,3 of row M=lane-16.  B 4x16 mirrored on columns.
    int half = lane >> 4;          // selects K pair within a 4-wide K step
    int lm = lane & 15;
    const float2* Arow0 = (const float2*)&Fl[wave][lm][0];       // points 0..15
    const float2* Arow1 = (const float2*)&Fl[wave][lm + 16][0];  // points 16..31
    int bidx = cbp[pc], gx = cxp[pc], gy = cyp[pc];

#pragma unroll
    for (int g = 0; g < 4; g++) {
        int ch = g * 16 + lm;
        const float2* Brow = (const float2*)&Wl[ch][0];
        v8f c0 = {};
        v8f c1 = {};
#pragma unroll
        for (int s = 0; s < 3; s++) {          // K = 12 in steps of 4
            int kk = s * 2 + half;             // float2 index: k_base = 4s + 2*half
            float2 t0 = Arow0[kk], t1 = Arow1[kk], tb = Brow[kk];
            v2f a0; a0.x = t0.x; a0.y = t0.y;
            v2f a1; a1.x = t1.x; a1.y = t1.y;
            v2f bb; bb.x = tb.x; bb.y = tb.y;
            c0 = __builtin_amdgcn_wmma_f32_16x16x4_f32(false, a0, false, bb,
                                                       (short)0, c0, false, false);
            c1 = __builtin_amdgcn_wmma_f32_16x16x4_f32(false, a1, false, bb,
                                                       (short)0, c1, false, false);
        }
        // C/D layout: VGPR r, lanes 0-15 -> row r, lanes 16-31 -> row r+8; col = lane%16.
        float sc = scaleArr[ch], sh = shiftArr[ch];
        float mmax = 0.f;  // ReLU output is >= 0
#pragma unroll
        for (int r = 0; r < 8; r++) {
            float v0 = fmaxf(c0[r] * sc + sh, 0.f);   // points (r + 8*half)
            float v1 = fmaxf(c1[r] * sc + sh, 0.f);   // points (16 + r + 8*half)
            mmax = fmaxf(mmax, fmaxf(v0, v1));
        }
        mmax = fmaxf(mmax, __shfl_xor(mmax, 16, 32));  // combine row halves
        if (half == 0 && p < P) {
            out[(((long)bidx * CCH + ch) * GRID_YD + gy) * GRID_XD + gx] = mmax;
        }
    }
}

extern "C" void kernel_launch(void* const* d_in, const int* in_sizes, int n_in,
                              void* d_out, int out_size, void* d_ws, size_t ws_size,
                              hipStream_t stream) {
    const float* pillars = (const float*)d_in[0];
    const float* W       = (const float*)d_in[1];
    const float* gamma   = (const float*)d_in[2];
    const float* beta    = (const float*)d_in[3];
    const int*   cx      = (const int*)d_in[4];
    const int*   cy      = (const int*)d_in[5];
    const int*   cb      = (const int*)d_in[6];
    const int*   np      = (const int*)d_in[7];
    int P = in_sizes[7];
    float* out = (float*)d_out;

    float* mom      = (float*)d_ws;   // 65 moment sums (padded to 72)
    float* scaleArr = mom + 72;       // 64 floats
    float* shiftArr = scaleArr + 64;  // 64 floats

    zero_kernel<<<2048, 256, 0, stream>>>(out, (long)out_size, mom);

    int momBlocks = 512;
    if (momBlocks > (P + 7) / 8) momBlocks = (P + 7) / 8;
    moments_kernel<<<momBlocks, 256, 0, stream>>>(pillars, cx, cy, np, mom, P);

    float invPN = 1.f / ((float)P * (float)NPTS);
    finalize_bn<<<1, 64, 0, stream>>>(mom, W, gamma, beta, scaleArr, shiftArr, invPN);

    int nb = (P + 7) / 8;
    pillar_wmma_kernel<<<nb, 256, 0, stream>>>(pillars, W, cx, cy, cb, np,
                                               scaleArr, shiftArr, out, P);
}